// GRUParallelDirection2d_3478923509771
// MI455X (gfx1250) — compile-verified
//
#include <hip/hip_runtime.h>
#include <stdint.h>

// GRU-style affine scan along W (contiguous axis). Memory-bound:
// read 512MiB + write 256MiB => ~35us floor at 23.3 TB/s; VALU ~1-2us.
// One lane per (b,c,h) row; W staged in 128-row x 32-step tiles into LDS via
// CDNA5 async global->LDS DMA (ASYNCcnt), double-buffered so the serial scan
// never exposes HBM latency. Async loads use the saddr (GVS) form: SGPR base
// + 32-bit VGPR byte offset => no per-tile 64-bit address VALU.

#define THREADS 128                       // 4 waves (wave32)
#define ROWS    128                       // one row per lane
#define T       32                        // timesteps per tile (128B/row/tile)
#define TP      36                        // padded LDS row stride: 144B, 16B-aligned,
                                          //   b128 reads hit all 64 banks exactly 2x (optimal)
#define WDIM    512
#define NT      (WDIM / T)                // 16 tiles
#define F4PR    (T / 4)                   // 8 float4 chunks per row per tile
#define LOADS   ((ROWS * F4PR) / THREADS) // 8 async b128 per thread per array

__device__ __forceinline__ uint32_t lds_off(const void* p) {
  // generic AS pointer to __shared__: low 32 bits = workgroup LDS byte offset
  return (uint32_t)(uintptr_t)p;
}

__device__ __forceinline__ void async_b128_saddr(uint32_t lds, int voff, const float* base) {
  // GLOBAL_LOAD_ASYNC_TO_LDS_B128, GVS mode: mem = SGPR64 + VGPR_I32.
  // Streaming data (each line consumed once) -> non-temporal hint.
  asm volatile("global_load_async_to_lds_b128 %0, %1, %2 th:TH_LOAD_NT"
               :
               : "v"(lds), "v"(voff), "s"(base)
               : "memory");
}

#define WAIT_ASYNCCNT(N) asm volatile("s_wait_asynccnt %0" ::"i"(N) : "memory")

__global__ __launch_bounds__(THREADS)
void GRUParallelDirection2d_scan_kernel(const float* __restrict__ z,
                                        const float* __restrict__ hc,
                                        const float* __restrict__ h0,
                                        float* __restrict__ out) {
  __shared__ alignas(16) float zs[2][ROWS * TP];
  __shared__ alignas(16) float hs[2][ROWS * TP];

  const int tid  = threadIdx.x;
  const int row0 = blockIdx.x * ROWS;           // global row base (max fits i32 math)

  // 8 consecutive lanes fetch one row's full 128B cacheline per async op.
  auto issue = [&](int tile, int buf) {
    const int t0 = tile * T;
#pragma unroll
    for (int j = 0; j < LOADS; ++j) {
      const int e  = j * THREADS + tid;
      const int r  = e >> 3;                    // F4PR == 8
      const int f4 = e & 7;
      const int gb = ((row0 + r) * WDIM + t0 + f4 * 4) * 4;  // byte offset < 2^31
      const int l  = r * TP + f4 * 4;
      async_b128_saddr(lds_off(&zs[buf][l]), gb, z);          // same voff VGPR,
      async_b128_saddr(lds_off(&hs[buf][l]), gb, hc);         // different saddr
    }
  };

  float hprev = h0[row0 + tid];                 // initial state, one scalar per row

  issue(0, 0);                                  // prime buffer 0
  for (int tile = 0; tile < NT; ++tile) {
    const int buf = tile & 1;
    if (tile + 1 < NT) {
      issue(tile + 1, buf ^ 1);                 // prefetch next tile into other buffer
      WAIT_ASYNCCNT(2 * LOADS);                 // tile's 16 ops done; next 16 in flight
    } else {
      WAIT_ASYNCCNT(0);
    }
    __syncthreads();                            // tile visible to all waves

    const float* zr = &zs[buf][tid * TP];
    const float* hr = &hs[buf][tid * TP];
    float* orow = out + (long long)(row0 + tid) * WDIM + tile * T;
#pragma unroll
    for (int q = 0; q < F4PR; ++q) {
      const float4 z4 = *(const float4*)(zr + q * 4);
      const float4 c4 = *(const float4*)(hr + q * 4);
      float4 o;
      // h = z*hc + (1-z)*h  ==  h += z*(hc - h)   (1 sub + 1 fma per step)
      hprev = fmaf(z4.x, c4.x - hprev, hprev); o.x = hprev;
      hprev = fmaf(z4.y, c4.y - hprev, hprev); o.y = hprev;
      hprev = fmaf(z4.z, c4.z - hprev, hprev); o.z = hprev;
      hprev = fmaf(z4.w, c4.w - hprev, hprev); o.w = hprev;
      *(float4*)(orow + q * 4) = o;             // lane writes a full 128B line per tile
    }
    __syncthreads();                            // reads done before buffer refill
  }
}

extern "C" void kernel_launch(void* const* d_in, const int* in_sizes, int n_in,
                              void* d_out, int out_size, void* d_ws, size_t ws_size,
                              hipStream_t stream) {
  const float* z  = (const float*)d_in[0];      // (B,C,H,W) f32
  const float* hc = (const float*)d_in[1];      // (B,C,H,W) f32
  const float* h0 = (const float*)d_in[2];      // (B,C,H,1) f32
  float* out = (float*)d_out;                   // (B,C,H,W) f32

  const int rows = in_sizes[2];                 // B*C*H = 131072
  dim3 grid(rows / ROWS);
  hipLaunchKernelGGL(GRUParallelDirection2d_scan_kernel, grid, dim3(THREADS), 0, stream,
                     z, hc, h0, out);
}